// TAGCN_12232066859621
// MI455X (gfx1250) — compile-verified
//
#include <hip/hip_runtime.h>
#include <hip/hip_bf16.h>
#include <math.h>

#define N_NODES 50000
#define N_EDGES 800000
#define F_DIM   128     // feature width propagated through hops (both layers)
#define H_OUT1  128
#define C_OUT   64
#define K_TOT   512     // (K_HOPS+1) * F_DIM, fused GEMM depth

typedef __attribute__((ext_vector_type(2))) float v2f;
typedef __attribute__((ext_vector_type(8))) float v8f;

// ---------------- elementwise helpers ----------------

__global__ void zero_f32_kernel(float* __restrict__ p, int n) {
    int i = blockIdx.x * blockDim.x + threadIdx.x;
    if (i < n) p[i] = 0.0f;
}

__global__ void degree_kernel(const int* __restrict__ col, float* __restrict__ deg) {
    int e = blockIdx.x * blockDim.x + threadIdx.x;
    if (e < N_EDGES) atomicAdd(&deg[col[e]], 1.0f);
}

__global__ void dinv_kernel(float* __restrict__ deg) {
    int i = blockIdx.x * blockDim.x + threadIdx.x;
    if (i < N_NODES) {
        float d = deg[i];
        deg[i] = (d > 0.0f) ? rsqrtf(d) : 0.0f;
    }
}

__global__ void norm_kernel(const int* __restrict__ row, const int* __restrict__ col,
                            const float* __restrict__ dinv, float* __restrict__ nrm) {
    int e = blockIdx.x * blockDim.x + threadIdx.x;
    if (e < N_EDGES) nrm[e] = dinv[row[e]] * dinv[col[e]];
}

// ---------------- sparse propagation (scatter-add) ----------------
// 32 lanes per edge; each lane handles a float4 chunk of the 128-feature row.
__global__ void spmm_scatter_kernel(const int* __restrict__ row, const int* __restrict__ col,
                                    const float* __restrict__ nrm,
                                    const float* __restrict__ hin, float* __restrict__ hout) {
    long long tid = (long long)blockIdx.x * blockDim.x + threadIdx.x;
    int e = (int)(tid >> 5);
    int f = ((int)tid & 31) * 4;
    if (e >= N_EDGES) return;
    float w = nrm[e];
    const float4 v = *(const float4*)(hin + (long long)row[e] * F_DIM + f);
    float* dst = hout + (long long)col[e] * F_DIM + f;
    atomicAdd(dst + 0, w * v.x);
    atomicAdd(dst + 1, w * v.y);
    atomicAdd(dst + 2, w * v.z);
    atomicAdd(dst + 3, w * v.w);
}

// ---------------- fused TAGConv GEMM via WMMA ----------------
// out[N x NOUT] = concat_K(s0,s1,s2,s3)[N x 512] @ W[512 x NOUT] + bias (+ReLU if MODE==0)
//
// Block = 256 threads = 8 waves. blockIdx.y selects one 16-column tile of W;
// its 512x16 slice (32 KB) is staged once into LDS by all threads, then each
// wave computes one 16x16 C tile (8 row tiles per block) with
// V_WMMA_F32_16X16X4_F32 over K=512, B fragments served from LDS.
template <int NOUT, int MODE>
__global__ __launch_bounds__(256) void tag_gemm_wmma_kernel(
    const float* __restrict__ s0, const float* __restrict__ s1,
    const float* __restrict__ s2, const float* __restrict__ s3,
    const float* __restrict__ W, const float* __restrict__ bias,
    float* __restrict__ out) {
    __shared__ float Bs[K_TOT * 16];   // [k][n] slice of W, 32 KB

    const int col0 = blockIdx.y * 16;

    // ---- stage W[:, col0:col0+16] into LDS (coalesced in 16-float runs) ----
    for (int i = threadIdx.x; i < K_TOT * 16; i += 256) {
        int k = i >> 4;
        int n = i & 15;
        Bs[i] = W[k * NOUT + col0 + n];
    }
    __syncthreads();

    const int wave   = threadIdx.x >> 5;
    const int lane   = threadIdx.x & 31;
    const int lane16 = lane & 15;
    const int hi     = lane >> 4;                 // 0: lanes 0-15, 1: lanes 16-31
    const int row0   = (blockIdx.x * 8 + wave) * 16;
    if (row0 >= N_NODES) return;                  // whole-wave exit, after barrier
    const int m = row0 + lane16;                  // A row owned by this lane

    const float* segs[4] = { s0, s1, s2, s3 };

    v8f c = { 0.f, 0.f, 0.f, 0.f, 0.f, 0.f, 0.f, 0.f };

#pragma unroll
    for (int s = 0; s < 4; ++s) {
        const float* __restrict__ arow = segs[s] + (long long)m * F_DIM;
        const float* __restrict__ bseg = Bs + s * F_DIM * 16;
        for (int kk = 0; kk < F_DIM; kk += 4) {
            // A fragment (16x4, f32): lanes 0-15 hold K=kk,kk+1 ; lanes 16-31 hold K=kk+2,kk+3
            v2f a;
            a.x = arow[kk + hi * 2 + 0];
            a.y = arow[kk + hi * 2 + 1];
            // B fragment (4x16, f32) from LDS: VGPR0 -> K=kk (lo) / kk+2 (hi); VGPR1 -> kk+1 / kk+3
            v2f b;
            b.x = bseg[(kk + hi * 2 + 0) * 16 + lane16];
            b.y = bseg[(kk + hi * 2 + 1) * 16 + lane16];
            c = __builtin_amdgcn_wmma_f32_16x16x4_f32(false, a, false, b, (short)0, c,
                                                      false, false);
        }
    }

    // Epilogue: C/D layout VGPR j: lanes 0-15 -> M=row0+j, lanes 16-31 -> M=row0+j+8, N=col0+lane16
    const int n    = col0 + lane16;
    const float bv = bias[n];
#pragma unroll
    for (int j = 0; j < 8; ++j) {
        int mm  = row0 + j + hi * 8;
        float v = c[j] + bv;
        if (MODE == 0) v = (v > 0.0f) ? v : 0.0f;   // ReLU for layer 1
        out[(long long)mm * NOUT + n] = v;
    }
}

// ---------------- row-wise log_softmax over C=64, in place ----------------
// One wave per row: 2 elements per lane, wave32 shuffle reductions.
__global__ void log_softmax_kernel(float* __restrict__ out) {
    int wave = threadIdx.x >> 5;
    int lane = threadIdx.x & 31;
    int r = blockIdx.x * (blockDim.x >> 5) + wave;
    if (r >= N_NODES) return;
    float* p = out + (long long)r * C_OUT;
    float a = p[lane];
    float b = p[lane + 32];
    float mx = fmaxf(a, b);
#pragma unroll
    for (int off = 16; off > 0; off >>= 1) mx = fmaxf(mx, __shfl_xor(mx, off, 32));
    float s = expf(a - mx) + expf(b - mx);
#pragma unroll
    for (int off = 16; off > 0; off >>= 1) s += __shfl_xor(s, off, 32);
    float lse = mx + logf(s);
    p[lane]      = a - lse;
    p[lane + 32] = b - lse;
}

// ---------------- launcher ----------------

extern "C" void kernel_launch(void* const* d_in, const int* in_sizes, int n_in,
                              void* d_out, int out_size, void* d_ws, size_t ws_size,
                              hipStream_t stream) {
    const float* x   = (const float*)d_in[0];
    const int*   ei  = (const int*)d_in[1];
    const float* W1  = (const float*)d_in[2];
    const float* b1  = (const float*)d_in[3];
    const float* W2  = (const float*)d_in[4];
    const float* b2  = (const float*)d_in[5];
    float*       out = (float*)d_out;

    const int* rowp = ei;             // edge_index[0]
    const int* colp = ei + N_EDGES;   // edge_index[1]

    // workspace layout (floats)
    float* ws   = (float*)d_ws;
    float* dinv = ws;                               // N
    float* nrm  = dinv + N_NODES;                   // E
    float* hb0  = nrm + N_EDGES;                    // N*128
    float* hb1  = hb0 + (size_t)N_NODES * F_DIM;    // N*128
    float* hb2  = hb1 + (size_t)N_NODES * F_DIM;    // N*128
    float* acc1 = hb2 + (size_t)N_NODES * F_DIM;    // N*128

    const int TB = 256;
    const int nodeBlocks = (N_NODES + TB - 1) / TB;
    const int edgeBlocks = (N_EDGES + TB - 1) / TB;
    const int featElems  = N_NODES * F_DIM;
    const int featBlocks = (featElems + TB - 1) / TB;
    const long long spmmThreads = (long long)N_EDGES * 32;
    const int spmmBlocks = (int)((spmmThreads + TB - 1) / TB);
    const int rowTiles   = N_NODES / 16;            // 3125, exact
    const int rowGroups  = (rowTiles + 7) / 8;      // 8 row tiles (waves) per block

    // --- gcn_norm ---
    zero_f32_kernel<<<nodeBlocks, TB, 0, stream>>>(dinv, N_NODES);
    degree_kernel<<<edgeBlocks, TB, 0, stream>>>(colp, dinv);
    dinv_kernel<<<nodeBlocks, TB, 0, stream>>>(dinv);
    norm_kernel<<<edgeBlocks, TB, 0, stream>>>(rowp, colp, dinv, nrm);

    // --- layer 1 hops: hb0 = A x, hb1 = A hb0, hb2 = A hb1 ---
    zero_f32_kernel<<<featBlocks, TB, 0, stream>>>(hb0, featElems);
    spmm_scatter_kernel<<<spmmBlocks, TB, 0, stream>>>(rowp, colp, nrm, x, hb0);
    zero_f32_kernel<<<featBlocks, TB, 0, stream>>>(hb1, featElems);
    spmm_scatter_kernel<<<spmmBlocks, TB, 0, stream>>>(rowp, colp, nrm, hb0, hb1);
    zero_f32_kernel<<<featBlocks, TB, 0, stream>>>(hb2, featElems);
    spmm_scatter_kernel<<<spmmBlocks, TB, 0, stream>>>(rowp, colp, nrm, hb1, hb2);

    // --- layer 1 fused GEMM (K=512) + bias + ReLU ---
    tag_gemm_wmma_kernel<H_OUT1, 0><<<dim3(rowGroups, H_OUT1 / 16), 256, 0, stream>>>(
        x, hb0, hb1, hb2, W1, b1, acc1);

    // --- layer 2 hops on acc1 ---
    zero_f32_kernel<<<featBlocks, TB, 0, stream>>>(hb0, featElems);
    spmm_scatter_kernel<<<spmmBlocks, TB, 0, stream>>>(rowp, colp, nrm, acc1, hb0);
    zero_f32_kernel<<<featBlocks, TB, 0, stream>>>(hb1, featElems);
    spmm_scatter_kernel<<<spmmBlocks, TB, 0, stream>>>(rowp, colp, nrm, hb0, hb1);
    zero_f32_kernel<<<featBlocks, TB, 0, stream>>>(hb2, featElems);
    spmm_scatter_kernel<<<spmmBlocks, TB, 0, stream>>>(rowp, colp, nrm, hb1, hb2);

    // --- layer 2 fused GEMM (K=512) + bias, straight into d_out ---
    tag_gemm_wmma_kernel<C_OUT, 1><<<dim3(rowGroups, C_OUT / 16), 256, 0, stream>>>(
        acc1, hb0, hb1, hb2, W2, b2, out);

    // --- log_softmax in place (8 rows per 256-thread block) ---
    log_softmax_kernel<<<N_NODES / 8, TB, 0, stream>>>(out);
}